// UniGINConv_18081812316775
// MI455X (gfx1250) — compile-verified
//
#include <hip/hip_runtime.h>
#include <hip/hip_bf16.h>
#include <math.h>

typedef float v2f __attribute__((ext_vector_type(2)));
typedef float v8f __attribute__((ext_vector_type(8)));

#define NN   100000
#define EE   50000
#define NNZ  1600000
#define DD   128

// ---------------------------------------------------------------------------
// Zero fill (graph-safe re-init of accumulators each call)
// ---------------------------------------------------------------------------
__global__ __launch_bounds__(256) void zero_kernel(float* __restrict__ p, long long n) {
    long long i = (long long)blockIdx.x * 256 + threadIdx.x;
    if (i < n) p[i] = 0.0f;
}

// ---------------------------------------------------------------------------
// Xp = X @ W  via V_WMMA_F32_16X16X4_F32 (full fp32 precision)
// block = 256 threads (8 wave32), one 16-row strip of Xp per block,
// wave w handles column tile n0 = 16*w. K-loop: 128/4 = 32 WMMA ops.
// A (16x4 f32): lanes 0-15 rows M=0..15; VGPR0 holds K = k0 + 2*half,
// VGPR1 holds K = k0 + 2*half + 1 (half = lane/16). B (4x16) mirrored.
// C/D (16x16 f32): VGPR r -> row M = r + 8*half, col N = lane%16.
// ---------------------------------------------------------------------------
__global__ __launch_bounds__(256) void gemm_wmma_f32(const float* __restrict__ X,
                                                     const float* __restrict__ W,
                                                     float* __restrict__ Xp) {
    const int strip = blockIdx.x;            // 16-row strip of X / Xp
    const int wave  = threadIdx.x >> 5;      // 0..7 -> column tile
    const int lane  = threadIdx.x & 31;
    const int m0    = strip * 16;
    const int n0    = wave * 16;
    const int rlane = lane & 15;             // row within tile (A), col (B/C/D)
    const int half  = lane >> 4;             // K-split half

    const float* xrow = X + (long long)(m0 + rlane) * DD;
    const int    ncol = n0 + rlane;

    v8f c = {};
#pragma unroll 8
    for (int k0 = 0; k0 < DD; k0 += 4) {
        const int ka = k0 + 2 * half;
        v2f a;
        a.x = xrow[ka];
        a.y = xrow[ka + 1];
        v2f b;
        b.x = W[(long long)ka * DD + ncol];
        b.y = W[(long long)(ka + 1) * DD + ncol];
        c = __builtin_amdgcn_wmma_f32_16x16x4_f32(
                /*neg_a=*/false, a, /*neg_b=*/false, b,
                /*c_mod=*/(short)0, c, /*reuse_a=*/false, /*reuse_b=*/false);
    }

#pragma unroll
    for (int r = 0; r < 8; ++r) {
        const int m = m0 + r + 8 * half;
        Xp[(long long)m * DD + ncol] = c[r];
    }
}

// ---------------------------------------------------------------------------
// Phase 1 scatter: Xe_sum[edges[i]] += Xp[vertex[i]];  cnt[edges[i]] += 1
// 256 threads = 2 nnz rows/block, 128 lanes per 512B row (coalesced gather).
// Destinations (25.6 MB + 0.2 MB) are L2-resident -> L2-rate fp32 atomics.
// ---------------------------------------------------------------------------
__global__ __launch_bounds__(256) void scatter_v2e(const float* __restrict__ Xp,
                                                   const int* __restrict__ vertex,
                                                   const int* __restrict__ edges,
                                                   float* __restrict__ XeSum,
                                                   float* __restrict__ cnt) {
    const int i = blockIdx.x * 2 + (threadIdx.x >> 7);
    if (i >= NNZ) return;
    const int d = threadIdx.x & 127;
    const int v = vertex[i];
    const int e = edges[i];
    const float val = Xp[(long long)v * DD + d];
    atomicAdd(&XeSum[(long long)e * DD + d], val);
    if (d == 0) atomicAdd(&cnt[e], 1.0f);
}

// ---------------------------------------------------------------------------
// Xe = Xe_sum / max(cnt, 1)   (in place in d_out's Xe output region)
// ---------------------------------------------------------------------------
__global__ __launch_bounds__(256) void edge_mean(float* __restrict__ Xe,
                                                 const float* __restrict__ cnt) {
    long long idx = (long long)blockIdx.x * 256 + threadIdx.x;
    if (idx >= (long long)EE * DD) return;
    const float c = cnt[idx >> 7];
    Xe[idx] = Xe[idx] / fmaxf(c, 1.0f);
}

// ---------------------------------------------------------------------------
// Phase 2 scatter: Xv[vertex[i]] += Xe[edges[i]]
// ---------------------------------------------------------------------------
__global__ __launch_bounds__(256) void scatter_e2v(const float* __restrict__ Xe,
                                                   const int* __restrict__ vertex,
                                                   const int* __restrict__ edges,
                                                   float* __restrict__ Xv) {
    const int i = blockIdx.x * 2 + (threadIdx.x >> 7);
    if (i >= NNZ) return;
    const int d = threadIdx.x & 127;
    const int v = vertex[i];
    const int e = edges[i];
    const float val = Xe[(long long)e * DD + d];
    atomicAdd(&Xv[(long long)v * DD + d], val);
}

// ---------------------------------------------------------------------------
// out = normalize((1+eps)*Xp + Xv) per row.
// 128 threads/row = 4 wave32; wave32 shfl_xor tree + LDS combine.
// ---------------------------------------------------------------------------
__global__ __launch_bounds__(128) void finalize_norm(const float* __restrict__ Xp,
                                                     const float* __restrict__ Xv,
                                                     const float* __restrict__ eps,
                                                     float* __restrict__ out) {
    const int v = blockIdx.x;
    const int d = threadIdx.x;
    const float scale = 1.0f + eps[0];
    const long long idx = (long long)v * DD + d;
    const float t = scale * Xp[idx] + Xv[idx];

    float s = t * t;
#pragma unroll
    for (int off = 16; off > 0; off >>= 1)
        s += __shfl_xor(s, off, 32);

    __shared__ float sm[4];
    const int w = d >> 5;
    if ((d & 31) == 0) sm[w] = s;
    __syncthreads();
    const float tot = sm[0] + sm[1] + sm[2] + sm[3];
    const float nrm = fmaxf(sqrtf(tot), 1e-12f);

    out[idx] = t / nrm;
}

// ---------------------------------------------------------------------------
// Launch
// inputs: 0:X [N,D] f32  1:W [D,D] f32  2:eps [1] f32  3:vertex [NNZ] i32
//         4:edges [NNZ] i32
// d_out:  [N*D] out  ++  [E*D] Xe
// d_ws :  Xp (N*D f32) | Xv (N*D f32) | cnt (E f32)
// ---------------------------------------------------------------------------
extern "C" void kernel_launch(void* const* d_in, const int* in_sizes, int n_in,
                              void* d_out, int out_size, void* d_ws, size_t ws_size,
                              hipStream_t stream) {
    const float* X      = (const float*)d_in[0];
    const float* W      = (const float*)d_in[1];
    const float* eps    = (const float*)d_in[2];
    const int*   vertex = (const int*)d_in[3];
    const int*   edges  = (const int*)d_in[4];

    float* out = (float*)d_out;
    float* Xe  = out + (long long)NN * DD;      // Xe output region (also Xe_sum accumulator)

    float* Xp  = (float*)d_ws;
    float* Xv  = Xp + (long long)NN * DD;
    float* cnt = Xv + (long long)NN * DD;

    const long long nd = (long long)NN * DD;    // 12.8M
    const long long ed = (long long)EE * DD;    // 6.4M

    // Re-zero accumulators every call (graph-replay safe).
    zero_kernel<<<(unsigned)((nd + 255) / 256), 256, 0, stream>>>(Xv, nd);
    zero_kernel<<<(unsigned)((ed + 255) / 256), 256, 0, stream>>>(Xe, ed);
    zero_kernel<<<(EE + 255) / 256, 256, 0, stream>>>(cnt, EE);

    // Xp = X @ W  (fp32 WMMA)
    gemm_wmma_f32<<<NN / 16, 256, 0, stream>>>(X, W, Xp);

    // vertex -> edge scatter-sum + counts
    scatter_v2e<<<NNZ / 2, 256, 0, stream>>>(Xp, vertex, edges, Xe, cnt);

    // mean
    edge_mean<<<(unsigned)((ed + 255) / 256), 256, 0, stream>>>(Xe, cnt);

    // edge -> vertex scatter-sum
    scatter_e2v<<<NNZ / 2, 256, 0, stream>>>(Xe, vertex, edges, Xv);

    // GIN residual + row L2 normalize
    finalize_norm<<<NN, 128, 0, stream>>>(Xp, Xv, eps, out);
}